// EdgeLoss_72679436583507
// MI455X (gfx1250) — compile-verified
//
#include <hip/hip_runtime.h>

#define IMG_B 16
#define IMG_H 256
#define IMG_W 256
#define BIG_DIST 1.0e6f

typedef __attribute__((ext_vector_type(2))) float v2f;
typedef __attribute__((ext_vector_type(8))) float v8f;
typedef __attribute__((address_space(1))) int as1_int;
typedef __attribute__((address_space(3))) int as3_int;

// ---------------------------------------------------------------------------
// Kernel 1: one block per (b,y) row.
//   - pred_edge  -> u8
//   - target_edge -> row scan -> g = 1D dist along W (capped BIG) -> g^2 (f32)
// ---------------------------------------------------------------------------
__device__ __forceinline__ bool bin_at(const float* __restrict__ p, int b, int y, int x) {
    return p[((size_t)b * IMG_H + y) * IMG_W + x] > 0.5f;
}

__global__ void EdgeLoss_edges_g2_kernel(const float* __restrict__ pred,
                                         const float* __restrict__ target,
                                         float* __restrict__ g2,
                                         unsigned char* __restrict__ pedge) {
    const int row = blockIdx.x;          // 0 .. B*H-1
    const int b = row >> 8;
    const int y = row & 255;
    const int x = threadIdx.x;           // 0 .. 255

    const bool tc = bin_at(target, b, y, x);
    const bool pc = bin_at(pred, b, y, x);

    bool t_off = false, p_off = false;
    #pragma unroll
    for (int dy = -1; dy <= 1; ++dy) {
        const int yy = y + dy;
        if (yy < 0 || yy >= IMG_H) continue;     // -inf padding: OOB never counts
        #pragma unroll
        for (int dx = -1; dx <= 1; ++dx) {
            const int xx = x + dx;
            if (xx < 0 || xx >= IMG_W) continue;
            t_off = t_off || !bin_at(target, b, yy, xx);
            p_off = p_off || !bin_at(pred, b, yy, xx);
        }
    }
    const bool tedge = tc && t_off;
    const bool pe    = pc && p_off;
    pedge[((size_t)b * IMG_H + y) * IMG_W + x] = pe ? 1u : 0u;

    // Row scan: nearest edge index to the left (max-scan) / right (min-scan).
    __shared__ int sL[IMG_W];
    __shared__ int sR[IMG_W];
    sL[x] = tedge ? x : -1;
    sR[x] = tedge ? x : (1 << 20);
    __syncthreads();
    #pragma unroll
    for (int off = 1; off < IMG_W; off <<= 1) {
        const int vL = (x >= off)          ? sL[x - off] : -1;
        const int vR = (x + off < IMG_W)   ? sR[x + off] : (1 << 20);
        __syncthreads();
        if (vL > sL[x]) sL[x] = vL;
        if (vR < sR[x]) sR[x] = vR;
        __syncthreads();
    }
    const int iL = sL[x];
    const int iR = sR[x];
    const float dL = (iL >= 0)    ? (float)(x - iL) : BIG_DIST;
    const float dR = (iR < IMG_W) ? (float)(iR - x) : BIG_DIST;
    const float g  = fminf(fminf(dL, dR), BIG_DIST);
    g2[((size_t)b * IMG_H + y) * IMG_W + x] = g * g;
}

// ---------------------------------------------------------------------------
// Kernel 2: one block per (b,x) column. Async-gather the strided g^2 column
// into LDS (CDNA5 global_load_async_to_lds path), brute-force the FH column
// min-plus, mask by pred_edge, deterministic block reduction -> partial[col].
// ---------------------------------------------------------------------------
__global__ void EdgeLoss_edt_col_kernel(const float* __restrict__ g2,
                                        const unsigned char* __restrict__ pedge,
                                        float* __restrict__ partial) {
    const int col = blockIdx.x;          // b*W + x
    const int b = col >> 8;
    const int x = col & 255;
    const int y = threadIdx.x;           // 0 .. 255

    __shared__ float cg2[IMG_H];
    __shared__ float red[IMG_H];

    const float* gp = g2 + ((size_t)b * IMG_H + y) * IMG_W + x;   // per-lane, stride W

    // CDNA5 async gather to LDS (per-lane global addr -> per-lane LDS addr),
    // tracked by ASYNCcnt. Builtin existence was proven by the round-1
    // type-check diagnostic, so call it unconditionally.
    __builtin_amdgcn_global_load_async_to_lds_b32(
        (as1_int*)(void*)gp, (as3_int*)&cg2[y], 0, 0);
    asm volatile("s_wait_asynccnt 0" ::: "memory");
    __syncthreads();

    float d2 = 3.0e38f;
    const float fy = (float)y;
    #pragma unroll 8
    for (int yp = 0; yp < IMG_H; ++yp) {
        const float dy = fy - (float)yp;
        d2 = fminf(d2, __fmaf_rn(dy, dy, cg2[yp]));
    }
    // g==0 on target-edge pixels => d2==0 there, so no extra mask needed.
    const float v = pedge[((size_t)b * IMG_H + y) * IMG_W + x] ? sqrtf(d2) : 0.0f;

    red[y] = v;
    __syncthreads();
    #pragma unroll
    for (int s = IMG_H / 2; s > 0; s >>= 1) {
        if (y < s) red[y] += red[y + s];
        __syncthreads();
    }
    if (y == 0) partial[col] = red[0];
}

// ---------------------------------------------------------------------------
// Kernel 3: single wave32. Sum 4096 partials with v_wmma_f32_16x16x4_f32:
// A = 16x4 chunk of partials, B = ones  =>  D[m][n] = sum_k A[m][k] + C[m][n].
// 64 accumulating WMMAs, then fold rows (8 acc VGPRs x lanes 0/16).
// A layout (16x4 f32): lane L holds M = L&15; VGPR0 = K = 2*(L>>4), VGPR1 = K+1.
// C/D layout (16x16 f32): lane L holds N = L&15; VGPR v holds M = v + 8*(L>>4).
// ---------------------------------------------------------------------------
__global__ void EdgeLoss_wmma_reduce_kernel(const float* __restrict__ partial,
                                            float* __restrict__ out,
                                            int n, float inv_count) {
    const int lane = threadIdx.x;        // 0 .. 31, exactly one wave, EXEC all-1s
    const int m  = lane & 15;
    const int k0 = (lane >> 4) << 1;

    v8f c = {0.f, 0.f, 0.f, 0.f, 0.f, 0.f, 0.f, 0.f};
    v2f ones;
    ones.x = 1.0f;
    ones.y = 1.0f;

    for (int base = 0; base < n; base += 64) {
        v2f a;
        a.x = partial[base + m * 4 + k0];
        a.y = partial[base + m * 4 + k0 + 1];
        c = __builtin_amdgcn_wmma_f32_16x16x4_f32(
                /*neg_a=*/false, a, /*neg_b=*/false, ones,
                /*c_mod=*/(short)0, c, /*reuse_a=*/false, /*reuse_b=*/false);
    }

    // lane holds column N = lane&15; its 8 acc regs are rows m0..m0+7.
    float s = c[0] + c[1] + c[2] + c[3] + c[4] + c[5] + c[6] + c[7];
    __shared__ float sh[32];
    sh[lane] = s;
    __syncthreads();
    if (lane == 0) {
        // column 0 total = rows 0..7 (lane 0) + rows 8..15 (lane 16)
        out[0] = (sh[0] + sh[16]) * inv_count;
    }
}

// ---------------------------------------------------------------------------
extern "C" void kernel_launch(void* const* d_in, const int* in_sizes, int n_in,
                              void* d_out, int out_size, void* d_ws, size_t ws_size,
                              hipStream_t stream) {
    const float* pred   = (const float*)d_in[0];
    const float* target = (const float*)d_in[1];

    const size_t npix = (size_t)IMG_B * IMG_H * IMG_W;     // 1,048,576
    unsigned char* ws = (unsigned char*)d_ws;
    float*         g2      = (float*)ws;                    // 4 MiB
    unsigned char* pedge   = ws + npix * sizeof(float);     // 1 MiB
    float*         partial = (float*)(ws + npix * sizeof(float) + npix); // 16 KiB

    const int nrows = IMG_B * IMG_H;                        // 4096
    const int ncols = IMG_B * IMG_W;                        // 4096

    EdgeLoss_edges_g2_kernel<<<nrows, IMG_W, 0, stream>>>(pred, target, g2, pedge);
    EdgeLoss_edt_col_kernel<<<ncols, IMG_H, 0, stream>>>(g2, pedge, partial);
    EdgeLoss_wmma_reduce_kernel<<<1, 32, 0, stream>>>(partial, (float*)d_out,
                                                      ncols, 1.0f / (float)npix);
}